// S6_49624052138283
// MI455X (gfx1250) — compile-verified
//
#include <hip/hip_runtime.h>

// S6-style fused kernel for MI455X (gfx1250, wave32, WMMA), single pass.
//
// Math (after simplifying the reference; the dA term is exactly 0):
//   s[m]   = sum_n (x@W2 + b2)[m,n] * (x@W3 + b3)[m,n]        (m = flat (b,l))
//   y[m,d] = x[m,d] * softplus((x@W1 + b1)[m,d]) * s[m]
//
// One block per 16-token tile. 8 waves; wave w computes 4 WMMA tiles of
// G = x@W1 (columns [64w,64w+64)) plus one 16x16 tile of the projection
// (waves 0-3: x@W2, waves 4-7: x@W3), all sharing the same A fragment.
// All matrix math runs on V_WMMA_F32_16X16X4_F32 (fp32 in/out, exact vs ref).

typedef __attribute__((ext_vector_type(2))) float v2f;
typedef __attribute__((ext_vector_type(8))) float v8f;
typedef __attribute__((ext_vector_type(4))) int   v4i;

#define DD 512   // model dim
#define NN 64    // state dim

#if defined(__gfx1250__) && __has_builtin(__builtin_amdgcn_global_load_async_to_lds_b128)
#define S6_HAVE_ASYNC_LDS 1
#else
#define S6_HAVE_ASYNC_LDS 0
#endif

__device__ __forceinline__ void s6_wait_async0() {
#if __has_builtin(__builtin_amdgcn_s_wait_asynccnt)
    __builtin_amdgcn_s_wait_asynccnt(0);
#else
    asm volatile("s_wait_asynccnt 0" ::: "memory");
#endif
}

__global__ __launch_bounds__(256) void s6_fused(
    const float* __restrict__ x,  const float* __restrict__ W1,
    const float* __restrict__ b1, const float* __restrict__ W2,
    const float* __restrict__ b2, const float* __restrict__ W3,
    const float* __restrict__ b3, float* __restrict__ y)
{
    __shared__ float xs[16 * DD];      // 32 KB: x tile, row-major
    __shared__ float pb[2 * 16 * NN];  //  8 KB: [Bm|Cm][m][n] projections
    __shared__ float ss[16];           // per-token scalar s[m]

    const int tile = blockIdx.x;
    const int tid  = threadIdx.x;
    const size_t base = (size_t)tile * 16 * DD;

    // ---- Stage 16x512 x-tile into LDS (2048 x b128 over 256 threads) ----
#if S6_HAVE_ASYNC_LDS
    {
        __attribute__((address_space(1))) v4i* gsrc =
            (__attribute__((address_space(1))) v4i*)(x + base);
        __attribute__((address_space(3))) v4i* ldst =
            (__attribute__((address_space(3))) v4i*)xs;
#pragma unroll
        for (int i = 0; i < 8; ++i) {
            const int idx = tid + 256 * i;            // 16-byte element index
            __builtin_amdgcn_global_load_async_to_lds_b128(gsrc + idx, ldst + idx, 0, 0);
        }
        s6_wait_async0();
    }
#else
    {
        const float4* xg = (const float4*)(x + base);
        float4* xl = (float4*)xs;
#pragma unroll
        for (int i = 0; i < 8; ++i) xl[tid + 256 * i] = xg[tid + 256 * i];
    }
#endif
    __syncthreads();

    const int lane = tid & 31;
    const int wave = tid >> 5;
    const int ll = lane & 15;          // M index (A) / N index (B,C,D)
    const int lh = lane >> 4;          // K-half selector
    const int col0 = wave * 64;        // this wave's W1 column block

    // Projection assignment: waves 0-3 -> W2 tile (w&3), waves 4-7 -> W3.
    const int which = wave >> 2;                   // 0: Bm, 1: Cm
    const int tp = wave & 3;                       // 16-col tile of projection
    const float* __restrict__ P = which ? W3 : W2;
    const int cp = tp * 16 + ll;                   // projection column

    v8f acc[4], accP;
#pragma unroll
    for (int r = 0; r < 8; ++r) accP[r] = 0.0f;
#pragma unroll
    for (int t = 0; t < 4; ++t)
#pragma unroll
        for (int r = 0; r < 8; ++r) acc[t][r] = 0.0f;

    for (int k = 0; k < DD; k += 4) {
        const int kk = k + 2 * lh;
        // A fragment from LDS: x[row=ll, K=kk..kk+1] (8B-aligned ds_load_b64)
        const v2f a = *(const v2f*)(xs + ll * DD + kk);
        // Projection tile (W2/W3 are 512x64 row-major, hot in L2).
        v2f bp; bp.x = P[(size_t)kk * NN + cp]; bp.y = P[(size_t)(kk + 1) * NN + cp];
        accP = __builtin_amdgcn_wmma_f32_16x16x4_f32(
            false, a, false, bp, (short)0, accP, false, false);
#pragma unroll
        for (int t = 0; t < 4; ++t) {
            const int c = col0 + t * 16 + ll;
            v2f b; b.x = W1[(size_t)kk * DD + c]; b.y = W1[(size_t)(kk + 1) * DD + c];
            acc[t] = __builtin_amdgcn_wmma_f32_16x16x4_f32(
                false, a, false, b, (short)0, acc[t], false, false);
        }
    }

    // ---- Projections -> LDS (bias applied), then s[m] = <Bm[m,:], Cm[m,:]> ----
    {
        const float bias = (which ? b3 : b2)[cp];
#pragma unroll
        for (int r = 0; r < 8; ++r)
            pb[which * (16 * NN) + (r + 8 * lh) * NN + cp] = accP[r] + bias;
    }
    __syncthreads();
    if (tid < 16) {
        float acc_s = 0.0f;
#pragma unroll 8
        for (int n = 0; n < NN; ++n)
            acc_s += pb[tid * NN + n] * pb[16 * NN + tid * NN + n];
        ss[tid] = acc_s;
    }
    __syncthreads();

    // ---- Epilogue: y = x * softplus(G + b1) * s ----
#pragma unroll
    for (int t = 0; t < 4; ++t) {
        const int c = col0 + t * 16 + ll;
        const float bias = b1[c];
#pragma unroll
        for (int r = 0; r < 8; ++r) {
            const int m = r + 8 * lh;                  // C/D layout row
            const float g = acc[t][r] + bias;
            const float dlt = (g > 20.0f) ? g : log1pf(__expf(g));
            y[base + (size_t)m * DD + c] = xs[m * DD + c] * dlt * ss[m];
        }
    }
}

// ---------------------------------------------------------------------------
extern "C" void kernel_launch(void* const* d_in, const int* in_sizes, int n_in,
                              void* d_out, int out_size, void* d_ws, size_t ws_size,
                              hipStream_t stream)
{
    const float* x  = (const float*)d_in[0];
    const float* W1 = (const float*)d_in[1];
    const float* b1 = (const float*)d_in[2];
    const float* W2 = (const float*)d_in[3];
    const float* b2 = (const float*)d_in[4];
    const float* W3 = (const float*)d_in[5];
    const float* b3 = (const float*)d_in[6];
    // d_in[7] ("A") is mathematically dead: dA is multiplied by 0 in the ref.

    float* y = (float*)d_out;
    (void)d_ws; (void)ws_size;

    const int M = in_sizes[0] / DD;          // B*L = 4096 tokens
    const int tiles = M / 16;                // 256 16-token tiles

    s6_fused<<<tiles, 256, 0, stream>>>(x, W1, b1, W2, b2, W3, b3, y);
}